// Net_2697239462706
// MI455X (gfx1250) — compile-verified
//
#include <hip/hip_runtime.h>
#include <math.h>

typedef __attribute__((ext_vector_type(2))) float v2f;
typedef __attribute__((ext_vector_type(8))) float v8f;

#define N_NODES   100000
#define NUM_FEAT  64
#define NUM_CLASS 40
#define N_EDGES   1600000
#define WPB       5             // waves per block (6250 tiles / 5 = 1250 blocks, no tail)
#define GEMM_BLOCK (WPB * 32)
#define ROW_TILES  (N_NODES / 16)          // 6250
#define GEMM_GRID  (ROW_TILES / WPB)       // 1250
#define KSTEPS     (NUM_FEAT / 4)          // 16
#define CTILES     3                       // 40 classes -> 3 x 16 (padded)

// Fragment-ordered weights: [step][which(l/r)][tile][lane] -> v2f
#define NW_V2F     (KSTEPS * 2 * CTILES * 32)      // 3072 v2f = 24 KB
// Manual LDS layout (float offsets) -- keeps all hot immediates < 64 KB.
#define OFF_W      0                               // 24 KB of weight fragments
#define OFF_OUT    (NW_V2F * 2)                    // 6144  : WPB * 16 * 48 floats
#define OFF_LSE    (OFF_OUT + WPB * 16 * 48)       // 9984  : WPB * 16 floats
#define OFF_X      (OFF_LSE + WPB * 16)            // 10064 : WPB * 16 * 64 floats
#define OFF_A      (OFF_X + WPB * 16 * NUM_FEAT)   // 15184 : WPB * 16 * 64 floats
#define SMEM_FLOATS (OFF_A + WPB * 16 * NUM_FEAT)  // 20304 floats = 81216 B

// ---------------------------------------------------------------------------
// Kernel 1: zero the scatter workspace (agg[N*F] and deg[N]) each call.
// ---------------------------------------------------------------------------
__global__ void zero_ws(float* __restrict__ agg, float* __restrict__ deg) {
  long i = (long)blockIdx.x * blockDim.x + threadIdx.x;
  const long tot = (long)N_NODES * NUM_FEAT;
  if (i < tot)     agg[i] = 0.0f;
  if (i < N_NODES) deg[i] = 0.0f;
}

// ---------------------------------------------------------------------------
// Kernel 2: edge scatter. 16 threads per edge, each thread moves a float4 of
// features: b128 gather from x[src] (L2-resident) + 4 global_atomic_add_f32
// into agg[dst]. Lane q==0 also counts degree.
// ---------------------------------------------------------------------------
__global__ void edge_scatter(const float* __restrict__ x,
                             const int*   __restrict__ index,
                             float* __restrict__ agg,
                             float* __restrict__ deg) {
  long gid = (long)blockIdx.x * blockDim.x + threadIdx.x;
  long e   = gid >> 4;
  int  q   = (int)(gid & 15);
  if (e >= N_EDGES) return;
  int src = index[e];
  int dst = index[N_EDGES + e];
  const float4 v = *(const float4*)(x + (long)src * NUM_FEAT + q * 4);
  float* a = agg + (long)dst * NUM_FEAT + q * 4;
  atomicAdd(a + 0, v.x);
  atomicAdd(a + 1, v.y);
  atomicAdd(a + 2, v.z);
  atomicAdd(a + 3, v.w);
  if (q == 0) atomicAdd(deg + dst, 1.0f);
}

// ---------------------------------------------------------------------------
// Kernel 3: fused mean-normalize + dual GEMM (WMMA f32 16x16x4) + bias
// + row-wise log-softmax.
//
// WMMA f32 16x16x4 fragment layouts (ISA 7.12.2):
//   A (16x4): lane = 16*(k/2) + m, vgpr = k%2  -> one aligned 8B read/lane
//   B (4x16): lane = 16*(k/2) + n, vgpr = k%2
//   D (16x16): M = r + 8*(lane/16), N = lane%16
// ---------------------------------------------------------------------------
__global__ void __launch_bounds__(GEMM_BLOCK)
sage_gemm(const float* __restrict__ x,
          const float* __restrict__ agg,
          const float* __restrict__ deg,
          const float* __restrict__ Wl,
          const float* __restrict__ bias,
          const float* __restrict__ Wr,
          float* __restrict__ out) {
  __shared__ __align__(16) float smem[SMEM_FLOATS];

  v2f*   sW   = (v2f*)smem;                 // fragment-ordered Wl/Wr
  float* sOut = smem + OFF_OUT;
  float* sLse = smem + OFF_LSE;

  const int tid  = threadIdx.x;
  const int wave = tid >> 5;
  const int lane = tid & 31;
  const int half = lane >> 4;   // k-group within fragment
  const int l16  = lane & 15;   // M (A) / N (B) index

  const long row0 = ((long)blockIdx.x * WPB + wave) * 16;

  // ---- Stage weights in fragment order: i = ((step*2+which)*3 + t)*32 + li
  for (int i = tid; i < NW_V2F; i += GEMM_BLOCK) {
    const int li    = i & 31;
    const int t     = (i >> 5) % CTILES;
    const int which = (i / (CTILES * 32)) & 1;
    const int step  = i / (2 * CTILES * 32);
    const int k     = 4 * step + 2 * (li >> 4);
    const int n     = t * 16 + (li & 15);
    const float* W  = which ? Wr : Wl;
    v2f w = {0.0f, 0.0f};
    if (n < NUM_CLASS) w = (v2f){W[k * NUM_CLASS + n], W[(k + 1) * NUM_CLASS + n]};
    sW[i] = w;
  }

  // ---- Stage this wave's 16x64 x / agg tiles with coalesced b128 loads ----
  {
    const float* xg = x   + row0 * NUM_FEAT;
    const float* ag = agg + row0 * NUM_FEAT;
    float* sx = smem + OFF_X + wave * (16 * NUM_FEAT);
    float* sa = smem + OFF_A + wave * (16 * NUM_FEAT);
    for (int i = lane; i < (16 * NUM_FEAT) / 4; i += 32) {
      float4 vx = *(const float4*)(xg + i * 4);
      float4 va = *(const float4*)(ag + i * 4);
      *(float4*)(sx + i * 4) = vx;
      *(float4*)(sa + i * 4) = va;
    }
  }
  __syncthreads();

  const float rdeg = 1.0f / fmaxf(deg[row0 + l16], 1.0f);

  const v2f* xf = (const v2f*)(smem + OFF_X + wave * (16 * NUM_FEAT) + l16 * NUM_FEAT);
  const v2f* af = (const v2f*)(smem + OFF_A + wave * (16 * NUM_FEAT) + l16 * NUM_FEAT);
  const v2f* wp = sW + lane;

  v8f acc0 = {}, acc1 = {}, acc2 = {};

#pragma unroll
  for (int step = 0; step < KSTEPS; ++step) {
    const v2f a_x   = xf[2 * step + half];
    const v2f a_agg = af[2 * step + half] * rdeg;    // fuse mean into A-load

    const int wb = step * (2 * CTILES * 32);         // v2f units
    const v2f bl0 = wp[wb];
    const v2f bl1 = wp[wb + 32];
    const v2f bl2 = wp[wb + 64];
    const v2f br0 = wp[wb + 96];
    const v2f br1 = wp[wb + 128];
    const v2f br2 = wp[wb + 160];

    acc0 = __builtin_amdgcn_wmma_f32_16x16x4_f32(false, a_agg, false, bl0, (short)0, acc0, false, false);
    acc1 = __builtin_amdgcn_wmma_f32_16x16x4_f32(false, a_agg, false, bl1, (short)0, acc1, false, false);
    acc2 = __builtin_amdgcn_wmma_f32_16x16x4_f32(false, a_agg, false, bl2, (short)0, acc2, false, false);
    acc0 = __builtin_amdgcn_wmma_f32_16x16x4_f32(false, a_x,   false, br0, (short)0, acc0, false, false);
    acc1 = __builtin_amdgcn_wmma_f32_16x16x4_f32(false, a_x,   false, br1, (short)0, acc1, false, false);
    acc2 = __builtin_amdgcn_wmma_f32_16x16x4_f32(false, a_x,   false, br2, (short)0, acc2, false, false);
  }

  // Bias per lane per column tile (pad columns get 0, they are discarded).
  const float b0 = bias[l16];
  const float b1 = bias[16 + l16];
  const float b2 = (32 + l16 < NUM_CLASS) ? bias[32 + l16] : 0.0f;

  float* so = sOut + wave * (16 * 48);
#pragma unroll
  for (int r = 0; r < 8; ++r) {
    const int M = r + 8 * half;
    so[M * 48 +      l16] = acc0[r] + b0;
    so[M * 48 + 16 + l16] = acc1[r] + b1;
    so[M * 48 + 32 + l16] = acc2[r] + b2;
  }
  __syncthreads();

  // Row-wise log-sum-exp (one row per lane 0..15).
  if (lane < 16) {
    const float* rowp = so + l16 * 48;
    float mx = rowp[0];
    for (int c = 1; c < NUM_CLASS; ++c) mx = fmaxf(mx, rowp[c]);
    float s = 0.0f;
    for (int c = 0; c < NUM_CLASS; ++c) s += __expf(rowp[c] - mx);
    sLse[wave * 16 + l16] = mx + __logf(s);
  }
  __syncthreads();

  // Coalesced write of log-softmax output.
  for (int i = lane; i < 16 * NUM_CLASS; i += 32) {
    const int r = i / NUM_CLASS, c = i % NUM_CLASS;
    out[(row0 + r) * NUM_CLASS + c] = so[r * 48 + c] - sLse[wave * 16 + r];
  }
}

// ---------------------------------------------------------------------------
extern "C" void kernel_launch(void* const* d_in, const int* in_sizes, int n_in,
                              void* d_out, int out_size, void* d_ws, size_t ws_size,
                              hipStream_t stream) {
  const float* x     = (const float*)d_in[0];
  const int*   index = (const int*)d_in[1];
  const float* Wl    = (const float*)d_in[2];
  const float* bl    = (const float*)d_in[3];
  const float* Wr    = (const float*)d_in[4];
  float* out = (float*)d_out;

  float* agg = (float*)d_ws;                          // N_NODES * NUM_FEAT f32
  float* deg = agg + (size_t)N_NODES * NUM_FEAT;      // N_NODES f32

  {
    const long tot = (long)N_NODES * NUM_FEAT;
    zero_ws<<<(int)((tot + 255) / 256), 256, 0, stream>>>(agg, deg);
  }
  {
    const long tot = (long)N_EDGES * 16;
    edge_scatter<<<(int)((tot + 255) / 256), 256, 0, stream>>>(x, index, agg, deg);
  }
  sage_gemm<<<GEMM_GRID, GEMM_BLOCK, 0, stream>>>(x, agg, deg, Wl, bl, Wr, out);
}